// MFVIWordBasedJointSYNAndSpanSRL_33904471834783
// MI455X (gfx1250) — compile-verified
//
#include <hip/hip_runtime.h>
#include <hip/hip_bf16.h>
#include <math.h>

// Problem constants (from reference: B=4, S=128)
#define BB 4
#define SSZ 128
#define S2 (SSZ * SSZ)
#define S3 (S2 * SSZ)
#define BSS (BB * S2)

typedef float v2f __attribute__((ext_vector_type(2)));
typedef float v8f __attribute__((ext_vector_type(8)));

// D = A(16x4 f32) * B(4x16 f32) + C(16x16 f32), wave32.
// Args: (neg_a, A, neg_b, B, c_mod, C, reuse_a, reuse_b)
#define WMMA4(Amat, Bmat, Cmat)                                               \
  __builtin_amdgcn_wmma_f32_16x16x4_f32(false, (Amat), false, (Bmat),         \
                                        (short)0, (Cmat), false, false)

// ---------------------------------------------------------------------------
// q = sigmoid(base), elementwise over the 3 concatenated (B,S,S) base arrays.
// ---------------------------------------------------------------------------
__global__ __launch_bounds__(256) void mfvi_sigmoid_kernel(
    const float* __restrict__ base, float* __restrict__ q, int n) {
  int t = blockIdx.x * blockDim.x + threadIdx.x;
  if (t < n) {
    float x = base[t];
    q[t] = 1.0f / (1.0f + expf(-x));
  }
}

// ---------------------------------------------------------------------------
// Main pass: one block per (b,i). 256 threads = 8 waves; wave w owns j-tile
// [16w, 16w+16). For each 4-wide k chunk: build masked 16x4 A tiles in the
// WMMA A layout (lane L: rows M = L&15, K = 2*(L>>4)+{0,1}; one b64 load per
// tensor per lane), B operands are q values replicated across N (contiguous
// float2 load in the matching K layout), and chain 8 f32 16x16x4 WMMAs onto
// 3 accumulators (base_b, base_e, base_s terms). Masks recomputed on the fly.
// ---------------------------------------------------------------------------
__global__ __launch_bounds__(256) void mfvi_main_kernel(
    const float* __restrict__ sibBE, const float* __restrict__ sibEB,
    const float* __restrict__ sibBB, const float* __restrict__ sibEE,
    const float* __restrict__ copB, const float* __restrict__ copE,
    const float* __restrict__ grdB, const float* __restrict__ grdE,
    const float* __restrict__ splt, const int* __restrict__ edge,
    const int* __restrict__ chartm, const float* __restrict__ qb,
    const float* __restrict__ qe, const float* __restrict__ qs,
    float* __restrict__ base_b, float* __restrict__ base_e,
    float* __restrict__ base_s) {
  const int bi = blockIdx.x;        // 0 .. B*S-1
  const int b = bi >> 7;            // / S
  const int i = bi & (SSZ - 1);
  const int lane = threadIdx.x & 31;
  const int wave = threadIdx.x >> 5;
  const int mrow = lane & 15;       // A-matrix row within tile
  const int half = lane >> 4;       // K-pair selector
  const int j0 = wave << 4;
  const int j = j0 + mrow;

  const int bS2 = b * S2;
  const int rowBI = bS2 + i * SSZ;        // [b,i,*] row in (B,S,S)
  const int base4 = bi * S2 + j * SSZ;    // [b,i,j,0] in (B,S,S,S)

  const bool em_ij = edge[rowBI + j] != 0;
  const bool c0_j = chartm[bS2 + j] != 0;                 // chart_mask[b,0,j]
  const int ip = (i == 0) ? (SSZ - 1) : (i - 1);          // roll(+1, axis=1)
  const bool chart_ij = chartm[bS2 + ip * SSZ + j] != 0;

  v8f accB = {}, accE = {}, accS = {};
  const v2f ones = {1.0f, 1.0f};

  for (int kc = 0; kc < SSZ; kc += 4) {
    const int k0 = kc + (half << 1);

    // 9 contiguous-k tensor tiles: one b64 load each.
    v2f tBE = *(const v2f*)(sibBE + base4 + k0);
    v2f tEB = *(const v2f*)(sibEB + base4 + k0);
    v2f tBB = *(const v2f*)(sibBB + base4 + k0);
    v2f tEE = *(const v2f*)(sibEE + base4 + k0);
    v2f tCB = *(const v2f*)(copB + base4 + k0);
    v2f tCE = *(const v2f*)(copE + base4 + k0);
    v2f tGB = *(const v2f*)(grdB + base4 + k0);
    v2f tGE = *(const v2f*)(grdE + base4 + k0);
    v2f tSP = *(const v2f*)(splt + base4 + k0);
    // transposed view s_grd_b[b,k,i,j] (L2-resident re-read; coalesced in j)
    float gbt[2] = {grdB[b * S3 + k0 * S2 + i * SSZ + j],
                    grdB[b * S3 + (k0 + 1) * S2 + i * SSZ + j]};

    // B-side q operands (depend on k only for fixed b,i): contiguous float2.
    v2f q_e_ik = *(const v2f*)(qe + rowBI + k0);
    v2f q_b_ik = *(const v2f*)(qb + rowBI + k0);
    v2f q_s_ik = *(const v2f*)(qs + rowBI + k0);
    v2f q_b_ki = {qb[bS2 + k0 * SSZ + i], qb[bS2 + (k0 + 1) * SSZ + i]};
    // A-folded q values (depend on j and k).
    v2f q_s_jk = *(const v2f*)(qs + bS2 + j * SSZ + k0);
    float qbkj[2] = {qb[bS2 + k0 * SSZ + j], qb[bS2 + (k0 + 1) * SSZ + j]};
    float qekj[2] = {qe[bS2 + k0 * SSZ + j], qe[bS2 + (k0 + 1) * SSZ + j]};
    float qskj[2] = {qs[bS2 + k0 * SSZ + j], qs[bS2 + (k0 + 1) * SSZ + j]};

    bool emk[2] = {edge[rowBI + k0] != 0, edge[rowBI + k0 + 1] != 0};
    bool c0k[2] = {chartm[bS2 + k0] != 0, chartm[bS2 + k0 + 1] != 0};

    v2f A_BE, A_BB, A_Bf, A_EB, A_EE, A_Ef, A_Sg, A_Sp;
#pragma unroll
    for (int c = 0; c < 2; ++c) {
      const int k = k0 + c;
      const bool em_jk = em_ij && emk[c];
      const bool mwm = em_jk && (i != k);                        // mask2o_wm
      const bool m2o = mwm && (j != k);                          // mask2o
      const bool mwmsw = em_jk && (i != j);                      // wm swapped
      const bool nc = mwm && (j <= k) && ((j >= i) || (k <= i)); // noncross
      const bool ncsw = mwmsw && (k <= j) && ((k >= i) || (j <= i));
      const int mx = (i > j) ? i : j;
      const bool span = chart_ij && (mx != k) && c0_j && c0k[c]; // span_m2o
      const bool outs = span && !((i <= k) && (j >= k));         // outside
      A_BE[c] = mwm ? tBE[c] : 0.0f;
      A_BB[c] = m2o ? tBB[c] : 0.0f;
      A_Bf[c] = (m2o ? tCB[c] : 0.0f) * qbkj[c] +
                (nc ? tGB[c] : 0.0f) * q_s_jk[c];
      A_EB[c] = mwmsw ? tEB[c] : 0.0f;
      A_EE[c] = m2o ? tEE[c] : 0.0f;
      A_Ef[c] = (m2o ? tCE[c] : 0.0f) * qekj[c] +
                (ncsw ? tGE[c] : 0.0f) * qskj[c];
      A_Sg[c] = outs ? gbt[c] : 0.0f;
      A_Sp[c] = span ? tSP[c] : 0.0f;
    }

    accB = WMMA4(A_BE, q_e_ik, accB);  // sib_be * qe[i,k]
    accB = WMMA4(A_BB, q_b_ik, accB);  // sib_bb * qb[i,k]
    accB = WMMA4(A_Bf, ones, accB);    // cop_b*qb[k,j] + grd_b*qs[j,k]
    accE = WMMA4(A_EB, q_b_ik, accE);  // sib_eb * qb[i,k]
    accE = WMMA4(A_EE, q_e_ik, accE);  // sib_ee * qe[i,k]
    accE = WMMA4(A_Ef, ones, accE);    // cop_e*qe[k,j] + grd_e*qs[k,j]
    accS = WMMA4(A_Sg, q_b_ki, accS);  // span_grd_b * qb[k,i]
    accS = WMMA4(A_Sp, q_s_ik, accS);  // split * qs[i,k]
  }

  // D layout: vgpr v, lane L -> M = v + 8*(L>>4); all N columns identical.
  if (mrow == 0) {
    const int jb = j0 + (half << 3);
#pragma unroll
    for (int v = 0; v < 8; ++v) {
      const int o = rowBI + jb + v;
      base_b[o] += accB[v];
      base_e[o] += accE[v];
      base_s[o] += accS[v];
    }
  }
}

// ---------------------------------------------------------------------------
// span_grd_e pass: base_s[b,i,j] += sum_k qe[k,j] * grd_e[b,k,j,i]*outside.
// One block per (b,j) so the lane dimension is i (contiguous loads). Runs
// after mfvi_main_kernel on the same stream -> no write races on base_s.
// ---------------------------------------------------------------------------
__global__ __launch_bounds__(256) void mfvi_spanE_kernel(
    const float* __restrict__ grdE, const int* __restrict__ chartm,
    const float* __restrict__ qe, float* __restrict__ base_s) {
  const int bj = blockIdx.x;
  const int b = bj >> 7;
  const int jj = bj & (SSZ - 1);
  const int lane = threadIdx.x & 31;
  const int wave = threadIdx.x >> 5;
  const int mrow = lane & 15;
  const int half = lane >> 4;
  const int i0 = wave << 4;
  const int i = i0 + mrow;

  const int bS2 = b * S2;
  const bool c0_j = chartm[bS2 + jj] != 0;
  const int ip = (i == 0) ? (SSZ - 1) : (i - 1);
  const bool chart_ij = chartm[bS2 + ip * SSZ + jj] != 0;
  const int mx = (i > jj) ? i : jj;

  v8f acc = {};
  for (int kc = 0; kc < SSZ; kc += 4) {
    const int k0 = kc + (half << 1);
    float g[2] = {grdE[b * S3 + k0 * S2 + jj * SSZ + i],
                  grdE[b * S3 + (k0 + 1) * S2 + jj * SSZ + i]};
    bool c0k[2] = {chartm[bS2 + k0] != 0, chartm[bS2 + k0 + 1] != 0};
    v2f Bq = {qe[bS2 + k0 * SSZ + jj], qe[bS2 + (k0 + 1) * SSZ + jj]};
    v2f A;
#pragma unroll
    for (int c = 0; c < 2; ++c) {
      const int k = k0 + c;
      const bool span = chart_ij && (mx != k) && c0_j && c0k[c];
      const bool outs = span && !((i <= k) && (jj >= k));
      A[c] = outs ? g[c] : 0.0f;
    }
    acc = WMMA4(A, Bq, acc);
  }

  if (mrow == 0) {
    const int ib = i0 + (half << 3);
#pragma unroll
    for (int v = 0; v < 8; ++v)
      base_s[bS2 + (ib + v) * SSZ + jj] += acc[v];
  }
}

// ---------------------------------------------------------------------------
extern "C" void kernel_launch(void* const* d_in, const int* in_sizes, int n_in,
                              void* d_out, int out_size, void* d_ws,
                              size_t ws_size, hipStream_t stream) {
  (void)in_sizes; (void)n_in; (void)out_size; (void)ws_size;
  const float* s_const = (const float*)d_in[0];
  const float* s_arg_begin = (const float*)d_in[1];
  const float* s_arg_end = (const float*)d_in[2];
  const float* sibBE = (const float*)d_in[3];
  const float* sibEB = (const float*)d_in[4];
  const float* sibBB = (const float*)d_in[5];
  const float* sibEE = (const float*)d_in[6];
  const float* copB = (const float*)d_in[7];
  const float* copE = (const float*)d_in[8];
  const float* grdB = (const float*)d_in[9];
  const float* grdE = (const float*)d_in[10];
  const float* splt = (const float*)d_in[11];
  const int* edge = (const int*)d_in[12];    // arg_begin_mask (bool)
  const int* chartm = (const int*)d_in[13];  // chart_mask (bool)

  // Output = (base_b, base_e, base_s) after 3 iterations; keep them in d_out.
  float* base_b = (float*)d_out;
  float* base_e = base_b + BSS;
  float* base_s = base_e + BSS;
  // q buffers in workspace (3 * 65536 floats = 768 KB).
  float* qb = (float*)d_ws;
  float* qe = qb + BSS;
  float* qs = qe + BSS;

  hipMemcpyAsync(base_b, s_arg_begin, BSS * sizeof(float),
                 hipMemcpyDeviceToDevice, stream);
  hipMemcpyAsync(base_e, s_arg_end, BSS * sizeof(float),
                 hipMemcpyDeviceToDevice, stream);
  hipMemcpyAsync(base_s, s_const, BSS * sizeof(float),
                 hipMemcpyDeviceToDevice, stream);

  for (int it = 0; it < 3; ++it) {
    mfvi_sigmoid_kernel<<<(3 * BSS + 255) / 256, 256, 0, stream>>>(
        base_b, qb, 3 * BSS);
    mfvi_main_kernel<<<BB * SSZ, 256, 0, stream>>>(
        sibBE, sibEB, sibBB, sibEE, copB, copE, grdB, grdE, splt, edge,
        chartm, qb, qe, qs, base_b, base_e, base_s);
    mfvi_spanE_kernel<<<BB * SSZ, 256, 0, stream>>>(grdE, chartm, qe, base_s);
  }
}